// PainnMessage_60189671686541
// MI455X (gfx1250) — compile-verified
//
#include <hip/hip_runtime.h>
#include <hip/hip_bf16.h>

#define N_NODES 50000
#define N_EDGES 400000
#define FDIM    128
#define NB      20
#define PI_OVER_CUTOFF 0.62831853071795864769f  // pi / 5.0

typedef float v2f __attribute__((ext_vector_type(2)));
typedef float v8f __attribute__((ext_vector_type(8)));

// ---------------------------------------------------------------------------
// f32 WMMA 16x16x4: D = A(16x4) * B(4x16) + C(16x16), wave32.
// ---------------------------------------------------------------------------
static __device__ __forceinline__ v8f wmma_f32_16x16x4(v2f a, v2f b, v8f c) {
#if __has_builtin(__builtin_amdgcn_wmma_f32_16x16x4_f32)
  return __builtin_amdgcn_wmma_f32_16x16x4_f32(false, a, false, b, (short)0, c,
                                               false, false);
#else
  asm volatile("v_wmma_f32_16x16x4_f32 %0, %1, %2, %0"
               : "+v"(c)
               : "v"(a), "v"(b));
  return c;
#endif
}

static __device__ __forceinline__ v8f splat8(float x) {
  v8f v = {x, x, x, x, x, x, x, x};
  return v;
}

// ---------------------------------------------------------------------------
// Kernel 1: S = silu(X @ W1 + b1) @ W2 + b2        (S -> d_ws, 50000x384 f32)
// One wave per 16-row tile. 3125 tiles exactly. 4 waves / block.
// ---------------------------------------------------------------------------
#define N_TILES 3125
#define HSTRIDE 132  // 128+4: lane bank stride 4 -> halves interleave, no conflicts

__global__ void __launch_bounds__(128)
painn_mlp_kernel(const float* __restrict__ X,
                 const float* __restrict__ W1, const float* __restrict__ b1,
                 const float* __restrict__ W2, const float* __restrict__ b2,
                 float* __restrict__ S) {
  __shared__ float Hs[4][16 * HSTRIDE];

  const int wave = threadIdx.x >> 5;
  const int lane = threadIdx.x & 31;
  const int tile = blockIdx.x * 4 + wave;
  const bool active = tile < N_TILES;

  const int n    = lane & 15;    // col-in-tile (B/C/D), row-in-tile (A)
  const int hi   = lane >> 4;    // lane half
  const int koff = hi << 1;      // A/B frag K sub-offset: 0 or 2
  float* __restrict__ H = Hs[wave];

  if (active) {
    // ---- GEMM1 + bias + SiLU, H (16x128) to LDS ----
    const float* __restrict__ xr = X + ((size_t)tile * 16 + n) * FDIM;
    #pragma unroll 2
    for (int ct = 0; ct < 8; ++ct) {
      const int cn = ct * 16 + n;
      v8f c = splat8(b1[cn]);  // bias pre-loaded into accumulator
      #pragma unroll 4
      for (int k0 = 0; k0 < FDIM; k0 += 4) {
        v2f a, b;
        a.x = xr[k0 + koff];
        a.y = xr[k0 + koff + 1];
        b.x = W1[(size_t)(k0 + koff) * FDIM + cn];
        b.y = W1[(size_t)(k0 + koff + 1) * FDIM + cn];
        c = wmma_f32_16x16x4(a, b, c);
      }
      #pragma unroll
      for (int r = 0; r < 8; ++r) {
        float x = c[r];
        // silu(x) = x * sigmoid(x), native v_rcp_f32 (no IEEE divide chain)
        float h = x * __builtin_amdgcn_rcpf(1.0f + __expf(-x));
        H[(r + hi * 8) * HSTRIDE + cn] = h;
      }
    }
  }
  __syncthreads();
  if (active) {
    // ---- GEMM2: S = H @ W2 + b2 ----
    #pragma unroll 2
    for (int ct = 0; ct < 24; ++ct) {
      const int cn = ct * 16 + n;
      v8f c = splat8(b2[cn]);
      #pragma unroll 4
      for (int k0 = 0; k0 < FDIM; k0 += 4) {
        v2f a, b;
        a.x = H[n * HSTRIDE + k0 + koff];
        a.y = H[n * HSTRIDE + k0 + koff + 1];
        b.x = W2[(size_t)(k0 + koff) * 384 + cn];
        b.y = W2[(size_t)(k0 + koff + 1) * 384 + cn];
        c = wmma_f32_16x16x4(a, b, c);
      }
      #pragma unroll
      for (int r = 0; r < 8; ++r) {
        S[((size_t)tile * 16 + r + hi * 8) * 384 + cn] = c[r];
      }
    }
  }
}

// ---------------------------------------------------------------------------
// Kernel 2: WMMA edge kernel. One wave processes 16 edges per group.
//   A (16x20): sine basis rows (per-edge), 5 k-steps of 4.
//   B (4x16) : W_filter tiles from LDS.
//   C tiles ft, ft+8, ft+16 computed together so each lane holds the matched
//   (gate_sv, gate_ev, msg) triple for feature f = ft*16 + (lane&15) across
//   8 edges (rows), then gathers S[src]/node_vector[src] and scatter-adds.
// ---------------------------------------------------------------------------
#define N_GROUPS (N_EDGES / 16)  // 25000, exact
#define WFS 400  // W_filter LDS row stride: 400%64==16 -> halves 32 banks apart

__global__ void __launch_bounds__(256)
painn_edge_kernel(const int* __restrict__ eidx,
                  const float* __restrict__ edist,
                  const float* __restrict__ ediff,
                  const float* __restrict__ nvec,
                  const float* __restrict__ Wf, const float* __restrict__ bf,
                  const float* __restrict__ S,
                  float* __restrict__ out_s,   // N_NODES x 128 (d_out)
                  float* __restrict__ out_v) { // N_NODES x 3 x 128
  __shared__ float wf_s[NB * WFS];
  __shared__ float bf_s[384];
  for (int i = threadIdx.x; i < NB * 384; i += blockDim.x)
    wf_s[(i / 384) * WFS + (i % 384)] = Wf[i];
  for (int i = threadIdx.x; i < 384; i += blockDim.x) bf_s[i] = bf[i];
  __syncthreads();

  const int lane    = threadIdx.x & 31;
  const int wave    = (blockIdx.x * (blockDim.x >> 5)) + (threadIdx.x >> 5);
  const int n_waves = gridDim.x * (blockDim.x >> 5);

  const int n    = lane & 15;  // A-row (edge-in-group) / C-col (mod 16)
  const int hi   = lane >> 4;
  const int koff = hi << 1;

  for (int g = wave; g < N_GROUPS; g += n_waves) {
    const int e   = g * 16 + n;        // edge owned by this lane (A rows)
    const int dst = eidx[2 * e];       // edge_idx[:,0]
    const int src = eidx[2 * e + 1];   // edge_idx[:,1]
    const float d    = edist[e];
    const float invd = __builtin_amdgcn_rcpf(d);
    const float ux = ediff[3 * e + 0] * invd;
    const float uy = ediff[3 * e + 1] * invd;
    const float uz = ediff[3 * e + 2] * invd;
    const float cut =
        (d < 5.0f) ? 0.5f * (__cosf(d * PI_OVER_CUTOFF) + 1.0f) : 0.0f;

    // A-fragment basis values this lane needs: n_basis = 4s+koff, 4s+koff+1
    float basA[10];
    #pragma unroll
    for (int s = 0; s < 5; ++s) {
      const int nb = 4 * s + koff;
      basA[2 * s]     = __sinf(d * (float)(nb + 1) * PI_OVER_CUTOFF) * invd;
      basA[2 * s + 1] = __sinf(d * (float)(nb + 2) * PI_OVER_CUTOFF) * invd;
    }

    // Broadcast per-edge scalars to the C-row dimension (rows hi*8 + r).
    float cutv[8], uxv[8], uyv[8], uzv[8];
    int srcv[8], dstv[8];
    #pragma unroll
    for (int r = 0; r < 8; ++r) {
      const int sl = hi * 8 + r;
      cutv[r] = __shfl(cut, sl, 32);
      uxv[r]  = __shfl(ux, sl, 32);
      uyv[r]  = __shfl(uy, sl, 32);
      uzv[r]  = __shfl(uz, sl, 32);
      srcv[r] = __shfl(src, sl, 32);
      dstv[r] = __shfl(dst, sl, 32);
    }

    #pragma unroll 1
    for (int ft = 0; ft < 8; ++ft) {
      const int f = ft * 16 + n;
      v8f csv = splat8(bf_s[f]);        // gate_state_vector cols
      v8f cev = splat8(bf_s[128 + f]);  // gate_edge_vector cols
      v8f cms = splat8(bf_s[256 + f]);  // message_scalar cols
      #pragma unroll
      for (int s = 0; s < 5; ++s) {
        const int k = 4 * s + koff;
        v2f a;
        a.x = basA[2 * s];
        a.y = basA[2 * s + 1];
        v2f b0, b1v, b2v;
        b0.x  = wf_s[k * WFS + f];
        b0.y  = wf_s[(k + 1) * WFS + f];
        b1v.x = wf_s[k * WFS + 128 + f];
        b1v.y = wf_s[(k + 1) * WFS + 128 + f];
        b2v.x = wf_s[k * WFS + 256 + f];
        b2v.y = wf_s[(k + 1) * WFS + 256 + f];
        csv = wmma_f32_16x16x4(a, b0, csv);
        cev = wmma_f32_16x16x4(a, b1v, cev);
        cms = wmma_f32_16x16x4(a, b2v, cms);
      }
      #pragma unroll
      for (int r = 0; r < 8; ++r) {
        const float cw = cutv[r];
        const float* __restrict__ Ssrc = S + (size_t)srcv[r] * 384;
        const float* __restrict__ nvs  = nvec + (size_t)srcv[r] * 384;
        const float gsv = csv[r] * cw * Ssrc[f];
        const float gev = cev[r] * cw * Ssrc[128 + f];
        const float ms  = cms[r] * cw * Ssrc[256 + f];
        float* __restrict__ os = out_s + (size_t)dstv[r] * 128;
        float* __restrict__ ov = out_v + (size_t)dstv[r] * 384;
        atomicAdd(os + f, ms);
        atomicAdd(ov + f,       fmaf(nvs[f],       gsv, gev * uxv[r]));
        atomicAdd(ov + 128 + f, fmaf(nvs[128 + f], gsv, gev * uyv[r]));
        atomicAdd(ov + 256 + f, fmaf(nvs[256 + f], gsv, gev * uzv[r]));
      }
    }
  }
}

// ---------------------------------------------------------------------------
extern "C" void kernel_launch(void* const* d_in, const int* in_sizes, int n_in,
                              void* d_out, int out_size, void* d_ws,
                              size_t ws_size, hipStream_t stream) {
  const int*   edge_idx    = (const int*)d_in[0];
  const float* edge_dist   = (const float*)d_in[1];
  const float* edge_diff   = (const float*)d_in[2];
  const float* node_scalar = (const float*)d_in[3];
  const float* node_vector = (const float*)d_in[4];
  const float* W_filter    = (const float*)d_in[5];
  const float* b_filter    = (const float*)d_in[6];
  const float* W1          = (const float*)d_in[7];
  const float* b1          = (const float*)d_in[8];
  const float* W2          = (const float*)d_in[9];
  const float* b2          = (const float*)d_in[10];

  float* out_scalar = (float*)d_out;                               // 50000*128
  float* out_vector = (float*)d_out + (size_t)N_NODES * FDIM;      // 50000*384
  float* S          = (float*)d_ws;                                // 50000*384

  // Residual form: seed outputs with the inputs.
  hipMemcpyAsync(out_scalar, node_scalar, (size_t)N_NODES * FDIM * sizeof(float),
                 hipMemcpyDeviceToDevice, stream);
  hipMemcpyAsync(out_vector, node_vector,
                 (size_t)N_NODES * 3 * FDIM * sizeof(float),
                 hipMemcpyDeviceToDevice, stream);

  // Node MLP via f32 WMMA.
  painn_mlp_kernel<<<(N_TILES + 3) / 4, 128, 0, stream>>>(node_scalar, W1, b1,
                                                          W2, b2, S);

  // WMMA edge filter + gather + L2-resident scatter-add.
  painn_edge_kernel<<<1024, 256, 0, stream>>>(edge_idx, edge_dist, edge_diff,
                                              node_vector, W_filter, b_filter,
                                              S, out_scalar, out_vector);
}